// CustomGRU_70540542869565
// MI455X (gfx1250) — compile-verified
//
#include <hip/hip_runtime.h>
#include <hip/hip_bf16.h>

// ---------------------------------------------------------------------------
// GRU forward for MI455X (gfx1250), fp32 math via V_WMMA_F32_16X16X4_F32.
//   Phase 0: pack weights into WMMA-native pair/quad layout (b64/b128 feeds)
//   Phase 1: input projections xz/xr/xh = x @ W^T + (w_b + r_b)   [WMMA GEMM]
//   Phase 2: persistent scan kernel, 8 WGs x 32 waves, h-tile lives in LDS
// ---------------------------------------------------------------------------

#define B_DIM 128
#define T_DIM 512
#define I_DIM 256
#define H_DIM 512
#define BT    (B_DIM * T_DIM)            // 65536
#define BTH   ((size_t)BT * H_DIM)       // 33554432 elements

typedef __attribute__((ext_vector_type(2))) float v2f;
typedef __attribute__((ext_vector_type(8))) float v8f;

__device__ __forceinline__ v8f wmma_f32(v2f a, v2f b, v8f c) {
  // D = A(16x4) * B(4x16) + C(16x16), full fp32
  return __builtin_amdgcn_wmma_f32_16x16x4_f32(
      /*neg_a=*/false, a, /*neg_b=*/false, b,
      /*c_mod=*/(short)0, c, /*reuse_a=*/false, /*reuse_b=*/false);
}

__device__ __forceinline__ float sigmoid_fast(float x) {
  return 1.0f / (1.0f + __expf(-x));
}
__device__ __forceinline__ float tanh_fast(float x) {
  float e = __expf(2.0f * x);
  return 1.0f - 2.0f / (e + 1.0f);   // correct limits at +/- inf
}

// ---------------------------------------------------------------------------
// Pack one weight matrix src[H][K] (row-major) into WMMA pair layout:
//   dst2[k2 * H + n] = { src[n][2k2], src[n][2k2+1] }      (float2 view)
// Lanes of a WMMA B-operand then read one contiguous b64 each.
// ---------------------------------------------------------------------------
__global__ __launch_bounds__(256)
void gru_pack2_kernel(const float* __restrict__ src, float* __restrict__ dst,
                      int K) {
  __shared__ float tile[32][33];
  const int k0 = blockIdx.x * 32;               // K block
  const int n0 = blockIdx.y * 32;               // row (output-n) block
  const int tx = threadIdx.x, ty = threadIdx.y; // block (32, 8)
  for (int j = ty; j < 32; j += 8)
    tile[j][tx] = src[(size_t)(n0 + j) * K + (k0 + tx)];
  __syncthreads();
  float2* d2 = (float2*)dst;
  for (int j = ty; j < 16; j += 8) {
    const int k2 = (k0 >> 1) + j;
    float2 v;
    v.x = tile[tx][2 * j];
    v.y = tile[tx][2 * j + 1];
    d2[(size_t)k2 * H_DIM + (n0 + tx)] = v;
  }
}

// Pack two matrices interleaved (z and r recurrent weights):
//   dst4[k2 * H + n] = { A[n][2k2], A[n][2k2+1], B[n][2k2], B[n][2k2+1] }
__global__ __launch_bounds__(256)
void gru_pack4_kernel(const float* __restrict__ srcA,
                      const float* __restrict__ srcB,
                      float* __restrict__ dst, int K) {
  __shared__ float ta[32][33];
  __shared__ float tb[32][33];
  const int k0 = blockIdx.x * 32;
  const int n0 = blockIdx.y * 32;
  const int tx = threadIdx.x, ty = threadIdx.y;
  for (int j = ty; j < 32; j += 8) {
    ta[j][tx] = srcA[(size_t)(n0 + j) * K + (k0 + tx)];
    tb[j][tx] = srcB[(size_t)(n0 + j) * K + (k0 + tx)];
  }
  __syncthreads();
  float4* d4 = (float4*)dst;
  for (int j = ty; j < 16; j += 8) {
    const int k2 = (k0 >> 1) + j;
    float4 v;
    v.x = ta[tx][2 * j]; v.y = ta[tx][2 * j + 1];
    v.z = tb[tx][2 * j]; v.w = tb[tx][2 * j + 1];
    d4[(size_t)k2 * H_DIM + (n0 + tx)] = v;
  }
}

// ---------------------------------------------------------------------------
// Phase 1: input projections.  One block = 64 rows of x (staged in LDS),
// 8 waves cover 3 matrices x 32 n-tiles = 96 jobs, 4 M-subtiles per job
// share each B load.  pw* are packed-pair layout (see gru_pack2_kernel).
// ---------------------------------------------------------------------------
__global__ __launch_bounds__(256)
void gru_proj_kernel(const float* __restrict__ x,
                     const float* __restrict__ pwz, const float* __restrict__ pwr,
                     const float* __restrict__ pwh,
                     const float* __restrict__ w_bz, const float* __restrict__ w_br,
                     const float* __restrict__ w_bh,
                     const float* __restrict__ r_bz, const float* __restrict__ r_br,
                     const float* __restrict__ r_bh,
                     float* __restrict__ xz, float* __restrict__ xr,
                     float* __restrict__ xh) {
  __shared__ float xt[64][I_DIM + 2];           // padded to dodge bank conflicts
  const int m0  = blockIdx.x * 64;
  const int tid = threadIdx.x;

  // stage 64 x 256 tile of x (float4 loads)
  for (int i = tid; i < 64 * (I_DIM / 4); i += 256) {
    int row = i >> 6;                           // 64 float4 per row
    int c4  = (i & 63) << 2;
    float4 v = *(const float4*)&x[(size_t)(m0 + row) * I_DIM + c4];
    xt[row][c4 + 0] = v.x; xt[row][c4 + 1] = v.y;
    xt[row][c4 + 2] = v.z; xt[row][c4 + 3] = v.w;
  }
  __syncthreads();

  const int wave = tid >> 5, lane = tid & 31;
  const int lm = lane & 15;                     // A row / C col within tile
  const int lh = lane >> 4;                     // lane-half
  const int koff = lh * 2;

  for (int job = wave; job < 96; job += 8) {
    const int mat   = job >> 5;                 // 0=z 1=r 2=h
    const int ntile = job & 31;
    const float2* pw = (const float2*)((mat == 0) ? pwz : (mat == 1) ? pwr : pwh);
    float* out      = (mat == 0) ? xz  : (mat == 1) ? xr  : xh;
    const float* wb = (mat == 0) ? w_bz : (mat == 1) ? w_br : w_bh;
    const float* rb = (mat == 0) ? r_bz : (mat == 1) ? r_br : r_bh;
    const int n = ntile * 16 + lm;

    v8f a0 = {0,0,0,0,0,0,0,0}, a1 = a0, a2 = a0, a3 = a0;
    #pragma unroll 4
    for (int kk = 0; kk < I_DIM; kk += 4) {
      const int kl = kk + koff;
      const int k2 = (kk >> 1) + lh;
      float2 w2 = pw[(size_t)k2 * H_DIM + n];   // one b64: {B[kl][n], B[kl+1][n]}
      v2f b; b.x = w2.x; b.y = w2.y;
      v2f a;
      a.x = xt[lm][kl];      a.y = xt[lm][kl + 1];      a0 = wmma_f32(a, b, a0);
      a.x = xt[16 + lm][kl]; a.y = xt[16 + lm][kl + 1]; a1 = wmma_f32(a, b, a1);
      a.x = xt[32 + lm][kl]; a.y = xt[32 + lm][kl + 1]; a2 = wmma_f32(a, b, a2);
      a.x = xt[48 + lm][kl]; a.y = xt[48 + lm][kl + 1]; a3 = wmma_f32(a, b, a3);
    }
    const float bias = wb[n] + rb[n];
    #pragma unroll
    for (int v = 0; v < 8; ++v) {
      const int mrow = v + lh * 8;              // C/D layout: M = vgpr + 8*half
      out[(size_t)(m0 +  0 + mrow) * H_DIM + n] = a0[v] + bias;
      out[(size_t)(m0 + 16 + mrow) * H_DIM + n] = a1[v] + bias;
      out[(size_t)(m0 + 32 + mrow) * H_DIM + n] = a2[v] + bias;
      out[(size_t)(m0 + 48 + mrow) * H_DIM + n] = a3[v] + bias;
    }
  }
}

// ---------------------------------------------------------------------------
// Phase 2: persistent recurrent scan.  8 blocks x 1024 threads (32 waves).
// Block owns 16 batch rows; wave owns one 16-wide N-tile.  h lives in LDS.
// out[] arrives pre-filled with xh and is overwritten in-place with h_t.
// pzr = interleaved quad layout (Rz|Rr), ph = pair layout (Rh).
// ---------------------------------------------------------------------------
#define HPAD (H_DIM + 2)

__global__ __launch_bounds__(1024)
void gru_scan_kernel(const float* __restrict__ h0,
                     const float* __restrict__ pzr, const float* __restrict__ ph,
                     const float* __restrict__ xz, const float* __restrict__ xr,
                     float* __restrict__ out, float* __restrict__ hlast) {
  __shared__ float ht[16][HPAD];
  __shared__ float rh[16][HPAD];
  const int b0  = blockIdx.x * 16;
  const int tid = threadIdx.x;

  for (int i = tid; i < 16 * H_DIM; i += 1024) {
    int m = i >> 9, c = i & (H_DIM - 1);
    ht[m][c] = h0[(size_t)(b0 + m) * H_DIM + c];
  }
  __syncthreads();

  const int wave = tid >> 5, lane = tid & 31;
  const int lm = lane & 15, lh = lane >> 4;
  const int koff = lh * 2;
  const int n = wave * 16 + lm;                 // this wave's output column
  const float4* pzr4 = (const float4*)pzr;
  const float2* ph2  = (const float2*)ph;

  for (int t = 0; t < T_DIM; ++t) {
    // ---- sub-phase A: hz = h @ Rz^T, hr = h @ Rr^T -----------------------
    v8f accz = {0,0,0,0,0,0,0,0}, accr = accz;
    #pragma unroll 4
    for (int kk = 0; kk < H_DIM; kk += 4) {
      const int kl = kk + koff;
      const int k2 = (kk >> 1) + lh;
      v2f a;  a.x = ht[lm][kl];  a.y = ht[lm][kl + 1];
      float4 w4 = pzr4[(size_t)k2 * H_DIM + n];  // one b128 feeds two WMMAs
      v2f bz; bz.x = w4.x; bz.y = w4.y;
      v2f br; br.x = w4.z; br.y = w4.w;
      accz = wmma_f32(a, bz, accz);
      accr = wmma_f32(a, br, accr);
    }
    #pragma unroll
    for (int v = 0; v < 8; ++v) {
      const int m = v + lh * 8;
      const size_t g = ((size_t)(b0 + m) * T_DIM + t) * H_DIM + n;
      float z = sigmoid_fast(xz[g] + accz[v]);
      float r = sigmoid_fast(xr[g] + accr[v]);
      accz[v] = z;                               // keep z in registers
      rh[m][n] = r * ht[m][n];
    }
    __syncthreads();

    // ---- sub-phase B: hh = tanh(xh + (r.h) @ Rh^T); blend ----------------
    v8f acch = {0,0,0,0,0,0,0,0};
    #pragma unroll 4
    for (int kk = 0; kk < H_DIM; kk += 4) {
      const int kl = kk + koff;
      const int k2 = (kk >> 1) + lh;
      v2f a;  a.x = rh[lm][kl];  a.y = rh[lm][kl + 1];
      float2 w2 = ph2[(size_t)k2 * H_DIM + n];   // one b64
      v2f bh; bh.x = w2.x; bh.y = w2.y;
      acch = wmma_f32(a, bh, acch);
    }
    #pragma unroll
    for (int v = 0; v < 8; ++v) {
      const int m = v + lh * 8;
      const size_t g = ((size_t)(b0 + m) * T_DIM + t) * H_DIM + n;
      float hh = tanh_fast(out[g] + acch[v]);    // out[g] currently holds xh
      float z  = accz[v];
      float hn = (1.0f - z) * hh + z * ht[m][n];
      out[g]   = hn;                             // overwrite xh with h_t
      ht[m][n] = hn;
    }
    __syncthreads();                             // protect ht for next step
  }

  for (int i = tid; i < 16 * H_DIM; i += 1024) {
    int m = i >> 9, c = i & (H_DIM - 1);
    hlast[(size_t)(b0 + m) * H_DIM + c] = ht[m][c];
  }
}

// ---------------------------------------------------------------------------
extern "C" void kernel_launch(void* const* d_in, const int* in_sizes, int n_in,
                              void* d_out, int out_size, void* d_ws, size_t ws_size,
                              hipStream_t stream) {
  const float* x    = (const float*)d_in[0];
  const float* h0   = (const float*)d_in[1];
  const float* w_z  = (const float*)d_in[2];
  const float* w_r  = (const float*)d_in[3];
  const float* w_h  = (const float*)d_in[4];
  const float* r_z  = (const float*)d_in[5];
  const float* r_r  = (const float*)d_in[6];
  const float* r_h  = (const float*)d_in[7];
  const float* w_bz = (const float*)d_in[8];
  const float* w_br = (const float*)d_in[9];
  const float* w_bh = (const float*)d_in[10];
  const float* r_bz = (const float*)d_in[11];
  const float* r_br = (const float*)d_in[12];
  const float* r_bh = (const float*)d_in[13];

  float* out   = (float*)d_out;                  // [B,T,H] then [B,H]
  float* hlast = out + BTH;

  // workspace layout (floats)
  float* ws  = (float*)d_ws;
  float* xz  = ws;                               // BTH
  float* xr  = ws + BTH;                         // BTH
  float* pwz = ws + 2 * BTH;                     // I*H (packed pairs)
  float* pwr = pwz + (size_t)I_DIM * H_DIM;
  float* pwh = pwr + (size_t)I_DIM * H_DIM;
  float* pzr = pwh + (size_t)I_DIM * H_DIM;      // (H/2)*H*4 (packed quads)
  float* ph  = pzr + (size_t)(H_DIM / 2) * H_DIM * 4;  // (H/2)*H*2

  dim3 tb(32, 8);
  // Phase 0: pack weights into WMMA-native layouts
  gru_pack2_kernel<<<dim3(I_DIM / 32, H_DIM / 32), tb, 0, stream>>>(w_z, pwz, I_DIM);
  gru_pack2_kernel<<<dim3(I_DIM / 32, H_DIM / 32), tb, 0, stream>>>(w_r, pwr, I_DIM);
  gru_pack2_kernel<<<dim3(I_DIM / 32, H_DIM / 32), tb, 0, stream>>>(w_h, pwh, I_DIM);
  gru_pack4_kernel<<<dim3(H_DIM / 32, H_DIM / 32), tb, 0, stream>>>(r_z, r_r, pzr, H_DIM);
  gru_pack2_kernel<<<dim3(H_DIM / 32, H_DIM / 32), tb, 0, stream>>>(r_h, ph, H_DIM);

  // Phase 1: input projections (xh lands directly in d_out)
  gru_proj_kernel<<<BT / 64, 256, 0, stream>>>(
      x, pwz, pwr, pwh, w_bz, w_br, w_bh, r_bz, r_br, r_bh, xz, xr, out);

  // Phase 2: persistent recurrent scan
  gru_scan_kernel<<<B_DIM / 16, 1024, 0, stream>>>(
      h0, pzr, ph, xz, xr, out, hlast);
}